// UNetTFF_7404523618552
// MI455X (gfx1250) — compile-verified
//
#include <hip/hip_runtime.h>
#include <hip/hip_bf16.h>

typedef __attribute__((ext_vector_type(2))) float v2f;
typedef __attribute__((ext_vector_type(8))) float v8f;

// ---------------------------------------------------------------------------
// silu
// ---------------------------------------------------------------------------
__device__ __forceinline__ float silu_f(float g) {
    return g / (1.0f + __expf(-g));
}

// Raw workgroup split-barrier: no implicit vmem/loadcnt drain (unlike
// __syncthreads), so register prefetch loads may span it.  LDS visibility
// is handled explicitly with s_wait_asynccnt / s_wait_dscnt at the call site.
__device__ __forceinline__ void wg_barrier_relaxed() {
    asm volatile("s_barrier_signal -1\n\ts_barrier_wait -1" ::: "memory");
}

// LDS layout for the B tile: pair-interleaved, bs[p][c][2] holds
// B rows (2p, 2p+1) for block column c.  Pitch 160 floats (128 data + 32 pad)
// => lanes 0-15 and 16-31 of a ds_load_b64 hit disjoint bank halves.
#define BS_PITCH 160

// ---------------------------------------------------------------------------
// One 16-deep K-chunk.  Block tile = 256 rows x 64 cols, 8 waves,
// each wave owns two 16-row tiles (rows w*16 and 128 + w*16).
//  fast path (TAIL=false): B tile filled with GLOBAL_LOAD_ASYNC_TO_LDS_B32
//    (ASYNCcnt, no VGPR staging), A fragments prefetched during the fill,
//    raw split barriers, B fragments double-buffered in registers so each
//    s_wait_dscnt is covered by 8 WMMAs.
//  tail path: synchronous fill, K-overrun zeroed via cndmask; addresses are
//    clamped (never predicated) - out-of-range rows/cols only feed outputs
//    that are never stored, so EXEC stays all-1 for the WMMAs.
// ---------------------------------------------------------------------------
template <bool TAIL>
__device__ __forceinline__ void mm_chunk(
    const float* __restrict__ A, const float* __restrict__ Bm, float* bs,
    v8f acc[2][4], int k0, int K, int M, int ldb,
    long arowOff0, long arowOff1, int tm, int lane, int tid)
{
    // previous chunk's ds_loads are all waited-on by their WMMA uses,
    // so DScnt==0 here; the raw barrier is sufficient before refill.
    if (!TAIL) wg_barrier_relaxed();
    else       __syncthreads();

    if (!TAIL) {
        // ---- async fill: 512 pairs, 256 threads, 2 pairs each ----
#pragma unroll
        for (int t = 0; t < 2; ++t) {
            const int e = tid + t * 256;
            const int p = e >> 6;           // pair index 0..7 (rows 2p, 2p+1)
            const int c = e & 63;
            const int gc  = tm * 64 + c;
            const int gcc = gc < M ? gc : M - 1;     // clamp (unstored cols)
            const float* g0 = Bm + (long)(k0 + 2 * p) * ldb + gcc;
            const float* g1 = g0 + ldb;
            const unsigned l0 = (unsigned)(size_t)(bs + p * BS_PITCH + c * 2);
            asm volatile("global_load_async_to_lds_b32 %0, %1, off"
                         :: "v"(l0), "v"((unsigned long long)g0) : "memory");
            asm volatile("global_load_async_to_lds_b32 %0, %1, off"
                         :: "v"(l0 + 4u), "v"((unsigned long long)g1) : "memory");
        }
    } else {
        // ---- synchronous guarded fill ----
#pragma unroll
        for (int t = 0; t < 2; ++t) {
            const int e = tid + t * 256;
            const int p = e >> 6;
            const int c = e & 63;
            const int gc  = tm * 64 + c;
            const int gcc = gc < M ? gc : M - 1;
            const int kr0 = k0 + 2 * p;
            const int kr1 = kr0 + 1;
            const int kc0 = kr0 < K ? kr0 : K - 1;
            const int kc1 = kr1 < K ? kr1 : K - 1;
            float v0 = Bm[(long)kc0 * ldb + gcc];
            float v1 = Bm[(long)kc1 * ldb + gcc];
            v0 = (kr0 < K) ? v0 : 0.0f;              // true zero beyond K
            v1 = (kr1 < K) ? v1 : 0.0f;
            v2f pr; pr.x = v0; pr.y = v1;
            *(v2f*)(bs + p * BS_PITCH + c * 2) = pr;
        }
    }

    // ---- A-fragment prefetch for both row tiles (overlaps fill/barrier) ----
    const int half = lane >> 4;
    const int kb   = half << 1;
    v2f a[2][4];
#pragma unroll
    for (int ks = 0; ks < 4; ++ks) {
        const int ka = k0 + ks * 4 + kb;
        if (TAIL) {
            const int ka0 = ka < K ? ka : K - 1;     // clamp only; B side is 0
            const int ka1 = (ka + 1) < K ? (ka + 1) : K - 1;
            a[0][ks].x = A[arowOff0 + ka0];
            a[0][ks].y = A[arowOff0 + ka1];
            a[1][ks].x = A[arowOff1 + ka0];
            a[1][ks].y = A[arowOff1 + ka1];
        } else {
            a[0][ks].x = A[arowOff0 + ka];
            a[0][ks].y = A[arowOff0 + ka + 1];
            a[1][ks].x = A[arowOff1 + ka];
            a[1][ks].y = A[arowOff1 + ka + 1];
        }
    }

    if (!TAIL) {
        asm volatile("s_wait_asynccnt 0x0" ::: "memory");  // LDS tile landed
        wg_barrier_relaxed();
    } else {
        asm volatile("s_wait_dscnt 0x0" ::: "memory");     // ds_stores landed
        __syncthreads();
    }

    // ---- compute: B fragments double-buffered, 8 WMMA per K-step ----
    const int rc = lane & 15;
    v2f b[2][4];
#pragma unroll
    for (int j = 0; j < 4; ++j)
        b[0][j] = *(const v2f*)(bs + half * BS_PITCH + (j * 16 + rc) * 2);
#pragma unroll
    for (int ks = 0; ks < 4; ++ks) {
        const int cur = ks & 1, nxt = cur ^ 1;
        if (ks < 3) {
            const int basen = ((ks + 1) * 2 + half) * BS_PITCH;
#pragma unroll
            for (int j = 0; j < 4; ++j)
                b[nxt][j] = *(const v2f*)(bs + basen + (j * 16 + rc) * 2);
        }
#pragma unroll
        for (int j = 0; j < 4; ++j) {
            acc[0][j] = __builtin_amdgcn_wmma_f32_16x16x4_f32(
                false, a[0][ks], false, b[cur][j], (short)0, acc[0][j], false, false);
            acc[1][j] = __builtin_amdgcn_wmma_f32_16x16x4_f32(
                false, a[1][ks], false, b[cur][j], (short)0, acc[1][j], false, false);
        }
    }
}

// ---------------------------------------------------------------------------
// fp32 WMMA GEMM:  C[N,M] (op)= A[N,K] @ B[K,M]
//   mode 0: C = acc + bias           (bias may be null)
//   mode 1: C = C + rowscale[n*rsStride] * acc
//   mode 2: C = silu(acc) * C
//   mode 3: C = (acc + bias) * silu(C)
// Block = 256 threads = 8 waves; block tile = 256 rows x 64 cols
// (N is always a multiple of 256 for this model -> no row guards).
// ---------------------------------------------------------------------------
__global__ __launch_bounds__(256) void gemm_wmma_f32(
    const float* __restrict__ A, const float* __restrict__ B,
    const float* __restrict__ bias, const float* __restrict__ rowscale,
    float* __restrict__ C,
    int N, int K, int M, int lda, int ldb, int ldc, int rsStride, int mode)
{
    __shared__ float bs[8 * BS_PITCH];
    const int tid  = threadIdx.x;
    const int lane = tid & 31;
    const int wave = tid >> 5;
    const int tilesM = (M + 63) >> 6;
    const int tn = blockIdx.x / tilesM;
    const int tm = blockIdx.x % tilesM;

    const int half = lane >> 4;
    const int rc   = lane & 15;
    const int row0 = tn * 256 + wave * 16 + rc;
    const int row1 = row0 + 128;
    const long arowOff0 = (long)row0 * lda;
    const long arowOff1 = (long)row1 * lda;

    v8f acc[2][4];
#pragma unroll
    for (int r = 0; r < 2; ++r)
#pragma unroll
        for (int j = 0; j < 4; ++j) acc[r][j] = 0.0f;

    const int Kmain = K & ~15;
    int k0 = 0;
    for (; k0 < Kmain; k0 += 16)
        mm_chunk<false>(A, B, bs, acc, k0, K, M, ldb, arowOff0, arowOff1,
                        tm, lane, tid);
    if (k0 < K)
        mm_chunk<true>(A, B, bs, acc, k0, K, M, ldb, arowOff0, arowOff1,
                       tm, lane, tid);

#pragma unroll
    for (int r = 0; r < 2; ++r) {
#pragma unroll
        for (int j = 0; j < 4; ++j) {
            const int col = tm * 64 + j * 16 + rc;
            if (col >= M) continue;
            const float bv = bias ? bias[col] : 0.0f;
#pragma unroll
            for (int i = 0; i < 8; ++i) {
                const int row = tn * 256 + r * 128 + wave * 16 + half * 8 + i;
                const long idx = (long)row * ldc + col;
                const float a = acc[r][j][i];
                float v;
                if (mode == 0)      v = a + bv;
                else if (mode == 1) v = C[idx] + rowscale[(long)row * rsStride] * a;
                else if (mode == 2) v = silu_f(a) * C[idx];
                else                v = (a + bv) * silu_f(C[idx]);
                C[idx] = v;
            }
        }
    }
}

// ---------------------------------------------------------------------------
// Per-token router: softmax over 16 experts, greedy top-4, renormalize,
// scatter into dense [N,16] weight matrix (zeros elsewhere).
// ---------------------------------------------------------------------------
__global__ void route_topk_kernel(const float* __restrict__ logits,
                                  float* __restrict__ w, int N)
{
    const int n = blockIdx.x * blockDim.x + threadIdx.x;
    if (n >= N) return;
    float s[16];
    float mx = -1e30f;
#pragma unroll
    for (int e = 0; e < 16; ++e) { s[e] = logits[n * 16 + e]; mx = fmaxf(mx, s[e]); }
    float sum = 0.0f;
#pragma unroll
    for (int e = 0; e < 16; ++e) { s[e] = __expf(s[e] - mx); sum += s[e]; }
    const float inv = 1.0f / sum;
#pragma unroll
    for (int e = 0; e < 16; ++e) s[e] *= inv;

    float out[16];
#pragma unroll
    for (int e = 0; e < 16; ++e) out[e] = 0.0f;

    int   selIdx[4];
    float selVal[4];
    float vsum = 0.0f;
#pragma unroll
    for (int t = 0; t < 4; ++t) {
        int bi = 0; float bv = -1e30f;
#pragma unroll
        for (int e = 0; e < 16; ++e) {
            if (s[e] > bv) { bv = s[e]; bi = e; }
        }
        selIdx[t] = bi; selVal[t] = bv; vsum += bv;
        s[bi] = -1e30f;
    }
    const float invv = 1.0f / vsum;
#pragma unroll
    for (int t = 0; t < 4; ++t) out[selIdx[t]] = selVal[t] * invv;
#pragma unroll
    for (int e = 0; e < 16; ++e) w[n * 16 + e] = out[e];
}

// ---------------------------------------------------------------------------
// Sequence-axis concat: out[b, 0:S1, :] = x[b], out[b, S1:S1+S2, :] = sk[b]
// ---------------------------------------------------------------------------
__global__ void concat_seq_kernel(const float* __restrict__ x,
                                  const float* __restrict__ sk,
                                  float* __restrict__ out,
                                  int S1, int S2, int d)
{
    const long total = (long)2 * (S1 + S2) * d;
    const long i = (long)blockIdx.x * blockDim.x + threadIdx.x;
    if (i >= total) return;
    const int dS = S1 + S2;
    const int  j = (int)(i % d);
    const long r = i / d;
    const int  s = (int)(r % dS);
    const int  b = (int)(r / dS);
    out[i] = (s < S1) ? x[((long)b * S1 + s) * d + j]
                      : sk[((long)b * S2 + (s - S1)) * d + j];
}

// ---------------------------------------------------------------------------
// Host orchestration
// ---------------------------------------------------------------------------
enum {
    LIN0_W, LIN0_B, SG0_W, SG0_B, GATE, W1, W3, W2,
    SH1, SH3, SH2, LIN1_W, LIN1_B, SG1_W, SG1_B,
    LIN2_W, LIN2_B, SG2_W, SG2_B, NPARAM
};

static const int ENC_DIMS[8] = {768, 662, 556, 450, 344, 238, 132, 27};
static const int DEC_DIMS[8] = {27, 132, 238, 344, 450, 556, 662, 768};

extern "C" void kernel_launch(void* const* d_in, const int* in_sizes, int n_in,
                              void* d_out, int out_size, void* d_ws, size_t ws_size,
                              hipStream_t stream)
{
    (void)in_sizes; (void)n_in; (void)out_size; (void)ws_size;

    const float* x = (const float*)d_in[0];
    const float* P[15][NPARAM];
    {
        int idx = 1;
        for (int b = 0; b < 15; ++b)
            for (int k = 0; k < NPARAM; ++k)
                P[b][k] = (const float*)d_in[idx++];
    }

    // Workspace carve (bump allocator, 256B aligned)
    char* ws = (char*)d_ws;
    size_t off = 0;
    auto alloc = [&](size_t bytes) -> float* {
        float* p = (float*)(ws + off);
        off += (bytes + 255) & ~(size_t)255;
        return p;
    };
    const int NMAX = 4096, DMAX = 768;
    float* XA = alloc((size_t)NMAX * DMAX * 4);        // stage outputs / bottleneck out
    float* XB = alloc((size_t)NMAX * DMAX * 4);        // decoder concat buffer
    float* L  = alloc((size_t)NMAX * DMAX * 4);        // linear pre-activation
    float* S  = alloc((size_t)NMAX * DMAX * 4);        // swiglu output / MoE input
    float* SH = alloc((size_t)NMAX * 2 * DMAX * 4);    // shared-expert hidden [N,2d]
    float* H  = alloc((size_t)NMAX * DMAX * 4);        // expert hidden [N,d]
    float* R  = alloc((size_t)NMAX * DMAX * 4);        // routed+shared accumulator
    float* GL = alloc((size_t)NMAX * 16 * 4);          // gate logits
    float* WT = alloc((size_t)NMAX * 16 * 4);          // dense top-4 weights
    float* SK[7];
    for (int i = 0; i < 7; ++i)
        SK[i] = alloc((size_t)512 * ENC_DIMS[i + 1] * 4);

    auto gemm = [&](const float* A, const float* B, const float* bias,
                    const float* rs, float* C,
                    int N, int K, int M, int lda, int ldb, int ldc,
                    int rsStride, int mode) {
        const int tilesM = (M + 63) / 64;
        const int blocks = (N / 256) * tilesM;   // N is always a multiple of 256
        gemm_wmma_f32<<<blocks, 256, 0, stream>>>(
            A, B, bias, rs, C, N, K, M, lda, ldb, ldc, rsStride, mode);
    };

    // One full _moeff block: xin[N,din] -> xout[N,d]
    auto moeff = [&](const float* xin, const float* const* p,
                     int N, int din, int d, float* xout) {
        // x = swiglu(x@lin0 + b0, sg0)
        gemm(xin, p[LIN0_W], p[LIN0_B], nullptr, L, N, din, d, din, d, d, 0, 0);
        gemm(L, p[SG0_W] + d, p[SG0_B] + d, nullptr, S, N, d, d, d, 2 * d, d, 0, 0);
        gemm(L, p[SG0_W],     p[SG0_B],     nullptr, S, N, d, d, d, 2 * d, d, 0, 3);
        // gate -> softmax -> top4 weights
        gemm(S, p[GATE], nullptr, nullptr, GL, N, d, 16, d, 16, 16, 0, 0);
        route_topk_kernel<<<(N + 255) / 256, 256, 0, stream>>>(GL, WT, N);
        // shared experts: R = (silu(x@sh1) * (x@sh3)) @ sh2
        gemm(S, p[SH3], nullptr, nullptr, SH, N, d, 2 * d, d, 2 * d, 2 * d, 0, 0);
        gemm(S, p[SH1], nullptr, nullptr, SH, N, d, 2 * d, d, 2 * d, 2 * d, 0, 2);
        gemm(SH, p[SH2], nullptr, nullptr, R, N, 2 * d, d, 2 * d, d, d, 0, 0);
        // routed experts: R += w[:,e] * ((silu(x@w1_e)*(x@w3_e)) @ w2_e)
        for (int e = 0; e < 16; ++e) {
            const size_t eo = (size_t)e * d * d;
            gemm(S, p[W3] + eo, nullptr, nullptr, H, N, d, d, d, d, d, 0, 0);
            gemm(S, p[W1] + eo, nullptr, nullptr, H, N, d, d, d, d, d, 0, 2);
            gemm(H, p[W2] + eo, nullptr, WT + e, R, N, d, d, d, d, d, 16, 1);
        }
        // x = swiglu(R@lin1 + b1, sg1)
        gemm(R, p[LIN1_W], p[LIN1_B], nullptr, L, N, d, d, d, d, d, 0, 0);
        gemm(L, p[SG1_W] + d, p[SG1_B] + d, nullptr, S, N, d, d, d, 2 * d, d, 0, 0);
        gemm(L, p[SG1_W],     p[SG1_B],     nullptr, S, N, d, d, d, 2 * d, d, 0, 3);
        // x = swiglu(x@lin2 + b2, sg2)  -> xout
        gemm(S, p[LIN2_W], p[LIN2_B], nullptr, L, N, d, d, d, d, d, 0, 0);
        gemm(L, p[SG2_W] + d, p[SG2_B] + d, nullptr, xout, N, d, d, d, 2 * d, d, 0, 0);
        gemm(L, p[SG2_W],     p[SG2_B],     nullptr, xout, N, d, d, d, 2 * d, d, 0, 3);
    };

    // ---------------- encoder (N = 512 tokens throughout) ----------------
    const float* cur = x;
    const int N0 = 512;
    for (int b = 0; b < 7; ++b) {
        moeff(cur, P[b], N0, ENC_DIMS[b], ENC_DIMS[b + 1], SK[b]);  // output == skip
        cur = SK[b];
    }
    // ---------------- bottleneck ----------------
    moeff(cur, P[7], N0, 27, 27, XA);
    // ---------------- decoder (sequence grows by 256 per block) ----------------
    int Scur = 256;
    const float* curd = XA;
    for (int i = 0; i < 7; ++i) {
        const int din = DEC_DIMS[i], dout = DEC_DIMS[i + 1];
        const int Snew = Scur + 256;
        const long total = (long)2 * Snew * din;
        concat_seq_kernel<<<(unsigned)((total + 255) / 256), 256, 0, stream>>>(
            curd, SK[6 - i], XB, Scur, 256, din);
        const int Nn = 2 * Snew;
        float* outp = (i == 6) ? (float*)d_out : XA;
        moeff(XB, P[8 + i], Nn, din, dout, outp);
        curd = XA;
        Scur = Snew;
    }
}